// GCN_5643587027244
// MI455X (gfx1250) — compile-verified
//
#include <hip/hip_runtime.h>

typedef __attribute__((ext_vector_type(2))) float v2f;
typedef __attribute__((ext_vector_type(8))) float v8f;

#define HDIM 64
#define CDIM 32

// ---------------- utility kernels ----------------

__global__ void fill_f32(float* p, int n, float v) {
    int i = blockIdx.x * blockDim.x + threadIdx.x;
    if (i < n) p[i] = v;
}

// layer 1: xw[node, f] += W1[feat, f]   (one thread per (nnz, f) pair)
__global__ void scatter_w1(const int* __restrict__ attr_node,
                           const int* __restrict__ attr_feat,
                           const float* __restrict__ W1,
                           float* __restrict__ xw, int nnz) {
    int gid = blockIdx.x * blockDim.x + threadIdx.x;
    int e = gid >> 6;          // / HDIM
    int f = gid & 63;          // % HDIM
    if (e >= nnz) return;
    int node = attr_node[e];
    int feat = attr_feat[e];
    atomicAdd(&xw[(long long)node * HDIM + f], W1[(long long)feat * HDIM + f]);
}

// in-degree accumulation (deg pre-filled with 1.0 = self loop)
__global__ void deg_edges(const int* __restrict__ dst, float* __restrict__ deg, int e_cnt) {
    int i = blockIdx.x * blockDim.x + threadIdx.x;
    if (i < e_cnt) atomicAdd(&deg[dst[i]], 1.0f);
}

__global__ void dinv_finalize(float* __restrict__ deg, int n) {
    int i = blockIdx.x * blockDim.x + threadIdx.x;
    if (i < n) {
        float v = deg[i];
        deg[i] = (v > 0.0f) ? rsqrtf(v) : 0.0f;
    }
}

// y[i,f] = dinv[i]^2 * x[i,f]   (self-loop term initializes the accumulator)
template <int F>
__global__ void self_loop_init(const float* __restrict__ dinv,
                               const float* __restrict__ x,
                               float* __restrict__ y, int n) {
    int gid = blockIdx.x * blockDim.x + threadIdx.x;
    if (gid >= n * F) return;
    int node = gid / F;
    float d = dinv[node];
    y[gid] = d * d * x[gid];
}

// y[dst,f] += dinv[src]*dinv[dst] * x[src,f]  (one thread per (edge, f))
template <int F>
__global__ void prop_edges(const int* __restrict__ src, const int* __restrict__ dst,
                           const float* __restrict__ dinv,
                           const float* __restrict__ x, float* __restrict__ y,
                           int e_cnt) {
    int gid = blockIdx.x * blockDim.x + threadIdx.x;
    int e = gid / F;
    int f = gid % F;
    if (e >= e_cnt) return;
    int s = src[e];
    int d = dst[e];
    float w = dinv[s] * dinv[d];
    atomicAdd(&y[(long long)d * F + f], w * x[(long long)s * F + f]);
}

// out[i,c] = dinv[i]^2 * hw[i,c] + b2[c]   (also initializes poisoned d_out)
__global__ void out_init(const float* __restrict__ dinv, const float* __restrict__ hw,
                         const float* __restrict__ b2, float* __restrict__ out, int n) {
    int gid = blockIdx.x * blockDim.x + threadIdx.x;
    if (gid >= n * CDIM) return;
    int node = gid >> 5;
    int c = gid & 31;
    float d = dinv[node];
    out[gid] = d * d * hw[gid] + b2[c];
}

// ---------------- WMMA GEMM: hw = relu(h) @ W2 ----------------
// h: [n, 64] f32, W2: [64, 32] f32 (row major), hw: [n, 32] f32
// One wave computes a 16x32 output tile via v_wmma_f32_16x16x4_f32.
__global__ void __launch_bounds__(256)
gemm_relu_wmma(const float* __restrict__ h, const float* __restrict__ W2,
               float* __restrict__ hw, int n) {
    __shared__ float w2s[HDIM * CDIM];  // 8 KB
    for (int i = threadIdx.x; i < HDIM * CDIM; i += blockDim.x) w2s[i] = W2[i];
    __syncthreads();

    int wave = threadIdx.x >> 5;  // 0..7
    int lane = threadIdx.x & 31;
    int rowBase = (blockIdx.x * 8 + wave) * 16;
    if (rowBase >= n) return;  // wave-uniform: EXEC stays all-1s for WMMA

    int m = lane & 15;   // A row within tile / B,C column within 16-tile
    int g = lane >> 4;   // half-wave group

    const float* hrow = h + (long long)(rowBase + m) * HDIM;

    v8f acc0 = {};  // columns 0..15
    v8f acc1 = {};  // columns 16..31

#pragma unroll
    for (int t = 0; t < HDIM / 4; ++t) {
        int k = t * 4 + g * 2;  // K = 4t + 2g + component
        v2f a, b0, b1;
        a.x = fmaxf(hrow[k], 0.0f);      // fused ReLU
        a.y = fmaxf(hrow[k + 1], 0.0f);
        b0.x = w2s[k * CDIM + m];
        b0.y = w2s[(k + 1) * CDIM + m];
        b1.x = w2s[k * CDIM + 16 + m];
        b1.y = w2s[(k + 1) * CDIM + 16 + m];
        acc0 = __builtin_amdgcn_wmma_f32_16x16x4_f32(false, a, false, b0,
                                                     (short)0, acc0, false, false);
        acc1 = __builtin_amdgcn_wmma_f32_16x16x4_f32(false, a, false, b1,
                                                     (short)0, acc1, false, false);
    }

    // D layout: VGPR r -> M = r + 8g, N = m
    int outM = rowBase + 8 * g;
#pragma unroll
    for (int r = 0; r < 8; ++r) {
        hw[(long long)(outM + r) * CDIM + m]      = acc0[r];
        hw[(long long)(outM + r) * CDIM + 16 + m] = acc1[r];
    }
}

// ---------------- launcher ----------------

extern "C" void kernel_launch(void* const* d_in, const int* in_sizes, int n_in,
                              void* d_out, int out_size, void* d_ws, size_t ws_size,
                              hipStream_t stream) {
    const int*   attr_idx = (const int*)d_in[0];   // [2, NNZ]
    const int*   edge_idx = (const int*)d_in[1];   // [2, E]
    // d_in[2] = n, d_in[3] = d : device scalars (unused; derived from sizes)
    const float* W1 = (const float*)d_in[4];       // [D, 64]
    const float* W2 = (const float*)d_in[5];       // [64, 32]
    const float* b2 = (const float*)d_in[6];       // [32]
    float*       out = (float*)d_out;              // [n, 32]

    const int nnz   = in_sizes[0] / 2;
    const int e_cnt = in_sizes[1] / 2;
    const int n     = out_size / CDIM;

    const int* attr_node = attr_idx;
    const int* attr_feat = attr_idx + nnz;
    const int* src = edge_idx;
    const int* dst = edge_idx + e_cnt;

    // workspace carve-up (floats): dinv[n] | xw[n*64] | h[n*64] | hw[n*32]
    float* ws   = (float*)d_ws;
    float* dinv = ws;
    float* xw   = dinv + n;
    float* hbuf = xw + (size_t)n * HDIM;
    float* hw   = hbuf + (size_t)n * HDIM;

    const int T = 256;
    auto blocks = [](long long work, int t) { return (int)((work + t - 1) / t); };

    // 1) layer-1 sparse feature transform: xw = scatter_add(W1 rows)
    fill_f32<<<blocks((long long)n * HDIM, T), T, 0, stream>>>(xw, n * HDIM, 0.0f);
    scatter_w1<<<blocks((long long)nnz * HDIM, T), T, 0, stream>>>(attr_node, attr_feat, W1, xw, nnz);

    // 2) gcn_norm: deg (with self-loop) -> dinv
    fill_f32<<<blocks(n, T), T, 0, stream>>>(dinv, n, 1.0f);
    deg_edges<<<blocks(e_cnt, T), T, 0, stream>>>(dst, dinv, e_cnt);
    dinv_finalize<<<blocks(n, T), T, 0, stream>>>(dinv, n);

    // 3) propagate 1: h = selfloop + edge scatter (ReLU fused into GEMM load)
    self_loop_init<HDIM><<<blocks((long long)n * HDIM, T), T, 0, stream>>>(dinv, xw, hbuf, n);
    prop_edges<HDIM><<<blocks((long long)e_cnt * HDIM, T), T, 0, stream>>>(src, dst, dinv, xw, hbuf, e_cnt);

    // 4) hw = relu(h) @ W2 via v_wmma_f32_16x16x4_f32
    int rowTiles = (n + 15) / 16;
    gemm_relu_wmma<<<(rowTiles + 7) / 8, 256, 0, stream>>>(hbuf, W2, hw, n);

    // 5) propagate 2 + bias
    out_init<<<blocks((long long)n * CDIM, T), T, 0, stream>>>(dinv, hw, b2, out, n);
    prop_edges<CDIM><<<blocks((long long)e_cnt * CDIM, T), T, 0, stream>>>(src, dst, dinv, hw, out, e_cnt);
}